// SwitchFocusedLossAdditive_80221399155356
// MI455X (gfx1250) — compile-verified
//
#include <hip/hip_runtime.h>
#include <cstdint>
#include <cstddef>

// Problem geometry from the reference: B=128, S=32768, C=4.
#define S_LEN     32768
#define SPLIT     4                    // segment-blocks per row -> 512 WGs
#define SEG_LEN   (S_LEN / SPLIT)      // 8192 positions per block
#define CHUNK     1024                 // positions per TDM chunk (16KB logits)
#define NCHUNK    (SEG_LEN / CHUNK)    // 8
#define WSEG      (SEG_LEN / 32)       // 256 bitmask words per segment
#define TOLW      5

typedef __attribute__((ext_vector_type(4))) unsigned int v4u;
typedef __attribute__((ext_vector_type(8))) int          v8i_t;
typedef __attribute__((ext_vector_type(4))) int          v4i_t;

// ---------------------------------------------------------------------------
// TDM: async 1-D f32 tile load (global -> LDS), 6-arg builtin
// (u32x4 g0, i32x8 g1, i32x4, i32x4, i32x8, i32 cpol).
// D# per CDNA5 ISA.
// ---------------------------------------------------------------------------
__device__ __forceinline__ void tdm_load_f32_1d(uint32_t lds_off, uint64_t gaddr,
                                                uint32_t nelem) {
  v4u g0;
  g0.x = 1u;                                             // count=1, user mode
  g0.y = lds_off;                                        // LDS byte address
  g0.z = (uint32_t)(gaddr & 0xFFFFFFFFull);              // global addr [31:0]
  g0.w = (uint32_t)((gaddr >> 32) & 0x01FFFFFFull)       // global addr [56:32]
       | (2u << 30);                                     // type = 2 ("image")
  v8i_t g1;
  g1[0] = (int)(2u << 16);                               // data_size = 4 bytes
  g1[1] = (int)((nelem & 0xFFFFu) << 16);                // tensor_dim0 [15:0]
  g1[2] = (int)(((nelem >> 16) & 0xFFFFu) | (1u << 16)); // td0 hi | tensor_dim1=1
  g1[3] = (int)((nelem & 0xFFFFu) << 16);                // tile_dim0 = nelem
  g1[4] = 1;                                             // tile_dim1 = 1
  g1[5] = (int)nelem;                                    // tensor_dim0_stride lo
  g1[6] = 0;
  g1[7] = 0;
  v4i_t z4 = {0, 0, 0, 0};
  v8i_t z8 = {0, 0, 0, 0, 0, 0, 0, 0};
  __builtin_amdgcn_tensor_load_to_lds(g0, g1, z4, z4, z8, 0);
}

// argmax(l0..l3) >= 2  under jnp first-max tie rule
__device__ __forceinline__ bool pred_is_switch(float4 v) {
  return fmaxf(v.z, v.w) > fmaxf(v.x, v.y);
}

__global__ void swl_init(float* wsf, float* farG, int* hasTrueG, int B) {
  const int i = threadIdx.x;
  if (i < 2) wsf[i] = 0.0f;
  if (i < B) { farG[i] = 0.0f; hasTrueG[i] = 0; }
}

__global__ void swl_fin(const float* wsf, const float* farG,
                        const int* hasTrueG, float* out, int B) {
  __shared__ float red[4];
  const int tid = threadIdx.x, lane = tid & 31, wv = tid >> 5;
  float f = (tid < B && hasTrueG[tid]) ? farG[tid] : 0.0f;
  for (int off = 16; off > 0; off >>= 1) f += __shfl_down(f, off, 32);
  if (lane == 0) red[wv] = f;
  __syncthreads();
  if (tid == 0)
    out[0] = (wsf[0] + red[0] + red[1] + red[2] + red[3]) / wsf[1];
}

__launch_bounds__(1024, 1)
__global__ void swl_main(const float* __restrict__ logits,
                         const int*   __restrict__ labels,
                         const int*   __restrict__ segm,
                         float*       __restrict__ wsf,      // [0]=loss, [1]=cnt
                         float*       __restrict__ farG,     // per-row far sum
                         int*         __restrict__ hasTrueG) // per-row any-true
{
  __shared__ __align__(16) float ldsLog[2][CHUNK * 4];       // 32 KB dbl buffer
  __shared__ uint32_t mPh[WSEG + 2], mTh[WSEG + 2], mSh[WSEG + 2]; // +halo words
  __shared__ uint32_t mV[WSEG];
  __shared__ float redF[32], redR[32];
  __shared__ int   redI[32];
  __shared__ int   sAnyTrue;

  const int tid  = threadIdx.x;
  const int lane = tid & 31;
  const int wv   = tid >> 5;
  const int row  = blockIdx.y;
  const int segBase = blockIdx.x * SEG_LEN;

  const size_t rowOff = (size_t)row * S_LEN;
  const uint64_t segLogBase =
      (uint64_t)(uintptr_t)(logits + (rowOff + (size_t)segBase) * 4);

  if (tid == 0) {
    sAnyTrue = 0;
    tdm_load_f32_1d((uint32_t)(uintptr_t)&ldsLog[0][0], segLogBase, CHUNK * 4);
  }

  float lossAcc = 0.0f;

  // ---- Phase A: TDM-stream logits, build bitmasks, accumulate weighted CE --
  for (int c = 0; c < NCHUNK; ++c) {
    if (tid == 0) {
      if (c + 1 < NCHUNK) {
        tdm_load_f32_1d((uint32_t)(uintptr_t)&ldsLog[(c + 1) & 1][0],
                        segLogBase + (uint64_t)(c + 1) * (CHUNK * 16),
                        CHUNK * 4);
        __builtin_amdgcn_s_wait_tensorcnt(1);  // oldest (chunk c) has landed
      } else {
        __builtin_amdgcn_s_wait_tensorcnt(0);
      }
    }
    __syncthreads();

    const size_t pos = rowOff + segBase + c * CHUNK + tid;
    const float4 v = ((const float4*)&ldsLog[c & 1][0])[tid];  // ds b128
    const int lab = labels[pos];
    const int sg  = segm[pos];

    const float m01 = fmaxf(v.x, v.y);
    const float m23 = fmaxf(v.z, v.w);
    const float bm  = fmaxf(m01, m23);
    const float zs  = __expf(v.x - bm) + __expf(v.y - bm) +
                      __expf(v.z - bm) + __expf(v.w - bm);
    const float logz = bm + __logf(zs);

    const bool valid = (lab != -100);
    const int  sl    = valid ? lab : 0;
    const float picked = (sl == 0) ? v.x : (sl == 1) ? v.y
                       : (sl == 2) ? v.z : v.w;
    const float wgt = (sl >= 2) ? 5.0f : 0.1f;   // CLASS_WEIGHTS {.1,.1,5,5}
    lossAcc += valid ? wgt * (logz - picked) : 0.0f;

    const uint32_t bp = (uint32_t)__ballot(m23 > m01);
    const uint32_t bt = (uint32_t)__ballot(lab >= 2);
    const uint32_t bs = (uint32_t)__ballot(sg > 0);
    const uint32_t bv = (uint32_t)__ballot(valid);
    if (lane == 0) {
      const int w = c * 32 + wv;
      mPh[w + 1] = bp; mTh[w + 1] = bt; mSh[w + 1] = bs; mV[w] = bv;
    }
    __syncthreads();
  }

  // ---- Halo words: 32 positions either side, recomputed from global -------
  if (wv == 0) {                                  // left halo
    const int hpos = segBase - 32 + lane;
    bool psw = false, tsw = false, ssw = false;
    if (hpos >= 0) {
      const float4 hv = *(const float4*)(logits + (rowOff + (size_t)hpos) * 4);
      psw = pred_is_switch(hv);
      tsw = labels[rowOff + hpos] >= 2;
      ssw = segm[rowOff + hpos] > 0;
    }
    const uint32_t bp = (uint32_t)__ballot(psw);
    const uint32_t bt = (uint32_t)__ballot(tsw);
    const uint32_t bs = (uint32_t)__ballot(ssw);
    if (lane == 0) { mPh[0] = bp; mTh[0] = bt; mSh[0] = bs; }
  } else if (wv == 1) {                           // right halo
    const int hpos = segBase + SEG_LEN + lane;
    bool psw = false, tsw = false, ssw = false;
    if (hpos < S_LEN) {
      const float4 hv = *(const float4*)(logits + (rowOff + (size_t)hpos) * 4);
      psw = pred_is_switch(hv);
      tsw = labels[rowOff + hpos] >= 2;
      ssw = segm[rowOff + hpos] > 0;
    }
    const uint32_t bp = (uint32_t)__ballot(psw);
    const uint32_t bt = (uint32_t)__ballot(tsw);
    const uint32_t bs = (uint32_t)__ballot(ssw);
    if (lane == 0) { mPh[WSEG + 1] = bp; mTh[WSEG + 1] = bt; mSh[WSEG + 1] = bs; }
  }
  __syncthreads();

  // ---- Phase B: windowed boolean adjustments, 32 positions per word -------
  float farAcc = 0.0f;
  int   cnt    = 0;
  if (tid < WSEG) {
    const uint32_t P  = mPh[tid + 1], Pm = mPh[tid], Pp = mPh[tid + 2];
    const uint32_t T  = mTh[tid + 1], Tm = mTh[tid], Tp = mTh[tid + 2];
    const uint32_t Sg = mSh[tid + 1], Sm = mSh[tid], Sp = mSh[tid + 2];
    const uint32_t V  = mV[tid];

    const uint32_t segp1 = (Sg << 1) | (Sm >> 31);   // seg[t-1]
    const uint32_t segp2 = (Sg << 2) | (Sm >> 30);   // seg[t-2]
    const uint32_t segn1 = (Sg >> 1) | (Sp << 31);   // seg[t+1]
    const uint32_t pen = segp1 | segp2;
    const uint32_t rew = Sg | segn1;

    uint32_t pnear = P, tnear = T;                   // window-any over +/-TOL
#pragma unroll
    for (int d = 1; d <= TOLW; ++d) {
      pnear |= (P << d) | (Pm >> (32 - d));
      pnear |= (P >> d) | (Pp << (32 - d));
      tnear |= (T << d) | (Tm >> (32 - d));
      tnear |= (T >> d) | (Tp << (32 - d));
    }

    lossAcc += 2.0f * (float)__popc(P & pen   & V)   // SEG_PENALTY
             - 1.0f * (float)__popc(P & rew   & V)   // SEG_REWARD
             - 2.0f * (float)__popc(T & pnear & V);  // PROX_REWARD
    farAcc   = 1.5f * (float)__popc(P & ~tnear & V); // FAR_PENALTY (row-gated)
    cnt      = __popc(V);
    if (T) atomicOr(&sAnyTrue, 1);
  }

  // ---- Phase C: block reduction + global accumulation ----------------------
  for (int off = 16; off > 0; off >>= 1) {
    lossAcc += __shfl_down(lossAcc, off, 32);
    farAcc  += __shfl_down(farAcc, off, 32);
    cnt     += __shfl_down(cnt, off, 32);
  }
  if (lane == 0) { redF[wv] = lossAcc; redR[wv] = farAcc; redI[wv] = cnt; }
  __syncthreads();
  if (wv == 0) {
    float v2 = redF[lane];
    float f2 = redR[lane];
    int   c2 = redI[lane];
    for (int off = 16; off > 0; off >>= 1) {
      v2 += __shfl_down(v2, off, 32);
      f2 += __shfl_down(f2, off, 32);
      c2 += __shfl_down(c2, off, 32);
    }
    if (lane == 0) {
      atomicAdd(&wsf[0], v2);
      atomicAdd(&wsf[1], (float)c2);
      atomicAdd(&farG[row], f2);
      if (sAnyTrue) atomicOr(&hasTrueG[row], 1);
    }
  }
}

extern "C" void kernel_launch(void* const* d_in, const int* in_sizes, int n_in,
                              void* d_out, int out_size, void* d_ws, size_t ws_size,
                              hipStream_t stream) {
  const float* logits = (const float*)d_in[0];
  const int*   labels = (const int*)d_in[1];
  const int*   seg    = (const int*)d_in[2];
  float* out = (float*)d_out;

  const int B = in_sizes[1] / S_LEN;   // 128

  float* wsf      = (float*)d_ws;      // [0]=loss sum, [1]=valid count
  float* farG     = wsf + 2;           // [B] per-row far-penalty sums
  int*   hasTrueG = (int*)(farG + B);  // [B] per-row any-true flags

  swl_init<<<1, 256, 0, stream>>>(wsf, farG, hasTrueG, B);
  swl_main<<<dim3(SPLIT, B), 1024, 0, stream>>>(logits, labels, seg,
                                                wsf, farG, hasTrueG);
  swl_fin<<<1, 128, 0, stream>>>(wsf, farG, hasTrueG, out, B);
}